// IndexAddInplace_50543175139910
// MI455X (gfx1250) — compile-verified
//
#include <hip/hip_runtime.h>
#include <hip/hip_bf16.h>

// Scatter-add: out = x; out[idx[i]] += src[i]  (rows of D=512 f32)
// Pure bandwidth-bound op (~0.125 FLOP/byte): no WMMA applies.
// gfx1250 paths used: global_load_b128 w/ NT temporal hint (stream src/x once),
// native no-return global_atomic_add_f32 at device scope (RMW in L2 atomic units).

// Native clang vector type (HIP's float4 is a class and is rejected by
// __builtin_nontemporal_load).
typedef float v4f __attribute__((ext_vector_type(4)));

__device__ __forceinline__ void atomic_add_f32_noret_dev(float* addr, float val) {
    // No-return FP32 atomic add, device scope. Tracked via STOREcnt; S_ENDPGM's
    // implicit wait-idle guarantees completion at kernel end.
    asm volatile("global_atomic_add_f32 %0, %1, off scope:SCOPE_DEV"
                 :
                 : "v"(addr), "v"(val)
                 : "memory");
}

// ---------------------------------------------------------------------------
// Kernel 1: out = x, 16B per thread. NT load (x read exactly once); regular
// store so destination lines are resident in L2 for the atomic phase.
// ---------------------------------------------------------------------------
__global__ __launch_bounds__(256) void IndexAdd_copy_kernel(
    const v4f* __restrict__ x, v4f* __restrict__ out, long long n4) {
    long long i = (long long)blockIdx.x * blockDim.x + threadIdx.x;
    if (i < n4) {
        out[i] = __builtin_nontemporal_load(&x[i]);
    }
}

// ---------------------------------------------------------------------------
// Kernel 2: scatter-add. blockDim = 256; D/4 lanes cover one row (128 lanes
// for D=512), so each block processes 256/(D/4) rows. Each thread streams a
// v4f of src (NT) and issues 4 device-scope f32 atomics into out.
// ---------------------------------------------------------------------------
__global__ __launch_bounds__(256) void IndexAdd_scatter_kernel(
    const float* __restrict__ src, const int* __restrict__ idx,
    float* __restrict__ out, int n_src, int d) {
    const int vec_per_row   = d >> 2;                       // v4f's per row (128)
    const int rows_per_blk  = blockDim.x / vec_per_row;     // 2 for D=512
    const int row  = blockIdx.x * rows_per_blk + (int)threadIdx.x / vec_per_row;
    const int lane = (int)threadIdx.x % vec_per_row;        // v4f index in row

    if (row >= n_src) return;

    const int dst_row = idx[row];

    const v4f* srow = (const v4f*)(src + (size_t)row * d);
    v4f s = __builtin_nontemporal_load(&srow[lane]);

    float* o = out + (size_t)dst_row * d + (size_t)lane * 4;
    atomic_add_f32_noret_dev(o + 0, s.x);
    atomic_add_f32_noret_dev(o + 1, s.y);
    atomic_add_f32_noret_dev(o + 2, s.z);
    atomic_add_f32_noret_dev(o + 3, s.w);
}

extern "C" void kernel_launch(void* const* d_in, const int* in_sizes, int n_in,
                              void* d_out, int out_size, void* d_ws, size_t ws_size,
                              hipStream_t stream) {
    const float* x   = (const float*)d_in[0];   // (N_DST, D) f32
    const int*   idx = (const int*)d_in[1];     // (N_SRC,)  int
    const float* src = (const float*)d_in[2];   // (N_SRC, D) f32
    float*       out = (float*)d_out;           // (N_DST, D) f32

    const long long n_x_elems   = in_sizes[0];                 // N_DST * D
    const long long n_src_rows  = in_sizes[1];                 // N_SRC
    const long long n_src_elems = in_sizes[2];                 // N_SRC * D
    const int d = (int)(n_src_elems / n_src_rows);             // 512

    // Phase 1: out = x  (v4f-vectorized)
    const long long n4 = n_x_elems >> 2;
    const int copy_grid = (int)((n4 + 255) / 256);
    IndexAdd_copy_kernel<<<copy_grid, 256, 0, stream>>>(
        (const v4f*)x, (v4f*)out, n4);

    // Phase 2: atomic scatter-add of src rows into out
    const int vec_per_row  = d >> 2;                // 128
    const int rows_per_blk = 256 / vec_per_row;     // 2
    const int scat_grid = (int)((n_src_rows + rows_per_blk - 1) / rows_per_blk);
    IndexAdd_scatter_kernel<<<scat_grid, 256, 0, stream>>>(
        src, idx, out, (int)n_src_rows, d);
}